// GConvFocus_18167711662710
// MI455X (gfx1250) — compile-verified
//
#include <hip/hip_runtime.h>
#include <hip/hip_bf16.h>

// GConvFocus: space-to-depth + 3x3 conv + bias + SiLU, recast as a 6x6
// stride-2 pad-2 conv over the original x, computed as implicit GEMM on the
// CDNA5 matrix pipe with V_WMMA_F32_16X16X4_F32 (exact f32 math).
//
// Shapes: x (16,3,640,640) f32, W (64,12,3,3) f32, b (64) f32
// out (16,64,320,320) f32.  K = 3*6*6 = 108.

typedef float v2f __attribute__((ext_vector_type(2)));
typedef float v8f __attribute__((ext_vector_type(8)));

#define Bb     16
#define Cin    3
#define Hh     640
#define Ww     640
#define OH     320
#define OW     320
#define CO     64
#define KK     108        // 3 * 6 * 6
#define SROWS  20         // 2*8 + 4 input rows staged per WG
#define SCOLS  68         // 2*32 + 4 input cols staged per WG
#define WLDS_OFF  0                       // 64*108 = 6912 floats
#define ALDS_OFF  (CO*KK)                 // 6912
#define ALDS_SZ   (Cin*SROWS*SCOLS)       // 4080 floats
#define BLDS_OFF  (ALDS_OFF + ALDS_SZ)    // 10992
#define LDS_TOT   (BLDS_OFF + CO)         // 11056 floats = 44.2 KB

// LDS offset (in floats) of A element for reduction index k, excluding the
// per-lane (row,col) base:  k = ci*36 + r*6 + s  ->  (ci*SROWS + r)*SCOLS + s
__host__ __device__ constexpr int AOFF(int k) {
    int ci = k / 36;
    int rm = k - ci * 36;
    int r  = rm / 6;
    int s  = rm - r * 6;
    return (ci * SROWS + r) * SCOLS + s;
}

__device__ __forceinline__ float silu(float v) {
    return v * (1.0f / (1.0f + __expf(-v)));
}

__global__ __launch_bounds__(256) void
GConvFocus_18167711662710_kernel(const float* __restrict__ x,
                                 const float* __restrict__ Wg,
                                 const float* __restrict__ bias,
                                 float* __restrict__ out) {
    __shared__ float lds[LDS_TOT];

    const int tid = threadIdx.x;
    const int bx  = blockIdx.x;
    const int owb = bx % 10;                // ow block (32 wide)
    const int ohb = (bx / 10) % 40;         // oh block (8 tall)
    const int bat = bx / 400;               // batch
    const int oh0 = ohb * 8;
    const int ow0 = owb * 32;

    // ---- Stage weights: Wlds[co][k], k = ci*36 + (2kh+di)*6 + (2kw+dj),
    //      gathered from original OIHW with c4 = ci + 3*(di + 2*dj). ----
    for (int i = tid; i < CO * KK; i += 256) {
        int co = i / KK;
        int k  = i - co * KK;
        int ci = k / 36;
        int rm = k - ci * 36;
        int r  = rm / 6;
        int s  = rm - r * 6;
        int kh = r >> 1, di = r & 1;
        int kw = s >> 1, dj = s & 1;
        int c4 = ci + 3 * (di + 2 * dj);
        lds[WLDS_OFF + i] = Wg[((co * 12 + c4) * 3 + kh) * 3 + kw];
    }

    // ---- Stage input strip: As[ci][row][col], input row = 2*oh0-2+row,
    //      input col = 2*ow0-2+col, zero-padded at borders. ----
    for (int i = tid; i < ALDS_SZ; i += 256) {
        int ci  = i / (SROWS * SCOLS);
        int rm  = i - ci * (SROWS * SCOLS);
        int row = rm / SCOLS;
        int col = rm - row * SCOLS;
        int gr  = 2 * oh0 - 2 + row;
        int gc  = 2 * ow0 - 2 + col;
        float v = 0.0f;
        if (gr >= 0 && gr < Hh && gc >= 0 && gc < Ww)
            v = x[((bat * Cin + ci) * Hh + gr) * Ww + gc];
        lds[ALDS_OFF + i] = v;
    }

    if (tid < CO) lds[BLDS_OFF + tid] = bias[tid];

    __syncthreads();

    // ---- Per-wave tile: 2 oh rows x 16 ow x 64 co ----
    const int wv  = tid >> 5;           // wave 0..7
    const int ln  = tid & 31;
    const int hi  = ln >> 4;            // lane half: K sub-offset 0 / 2
    const int mx  = ln & 15;            // A: M index, B/D: N index
    const int dxw = (wv & 1) * 16;      // wave ow offset within WG tile
    const int oy  = (wv >> 1) * 2;      // wave oh offset within WG tile

    // A fragment base addresses (float index into lds) for the two M-tiles.
    const float* ap0 = &lds[ALDS_OFF + (2 * (oy + 0)) * SCOLS + 2 * (dxw + mx)];
    const float* ap1 = &lds[ALDS_OFF + (2 * (oy + 1)) * SCOLS + 2 * (dxw + mx)];
    // B fragment base addresses for the 4 N-tiles.
    const float* wp0 = &lds[WLDS_OFF + ( 0 + mx) * KK + 2 * hi];
    const float* wp1 = &lds[WLDS_OFF + (16 + mx) * KK + 2 * hi];
    const float* wp2 = &lds[WLDS_OFF + (32 + mx) * KK + 2 * hi];
    const float* wp3 = &lds[WLDS_OFF + (48 + mx) * KK + 2 * hi];

    v8f acc[2][4];
#pragma unroll
    for (int t = 0; t < 2; ++t)
#pragma unroll
        for (int n = 0; n < 4; ++n)
            acc[t][n] = (v8f){0.f, 0.f, 0.f, 0.f, 0.f, 0.f, 0.f, 0.f};

#pragma unroll
    for (int k0 = 0; k0 < KK; k0 += 4) {
        // compile-time pair selected by lane half (one v_cndmask per step)
        const int ao = hi ? AOFF(k0 + 2) : AOFF(k0);
        v2f a0 = *(const v2f*)(ap0 + ao);
        v2f a1 = *(const v2f*)(ap1 + ao);
        v2f b0 = *(const v2f*)(wp0 + k0);
        v2f b1 = *(const v2f*)(wp1 + k0);
        v2f b2 = *(const v2f*)(wp2 + k0);
        v2f b3 = *(const v2f*)(wp3 + k0);

        acc[0][0] = __builtin_amdgcn_wmma_f32_16x16x4_f32(false, a0, false, b0, (short)0, acc[0][0], false, false);
        acc[0][1] = __builtin_amdgcn_wmma_f32_16x16x4_f32(false, a0, false, b1, (short)0, acc[0][1], false, false);
        acc[0][2] = __builtin_amdgcn_wmma_f32_16x16x4_f32(false, a0, false, b2, (short)0, acc[0][2], false, false);
        acc[0][3] = __builtin_amdgcn_wmma_f32_16x16x4_f32(false, a0, false, b3, (short)0, acc[0][3], false, false);
        acc[1][0] = __builtin_amdgcn_wmma_f32_16x16x4_f32(false, a1, false, b0, (short)0, acc[1][0], false, false);
        acc[1][1] = __builtin_amdgcn_wmma_f32_16x16x4_f32(false, a1, false, b1, (short)0, acc[1][1], false, false);
        acc[1][2] = __builtin_amdgcn_wmma_f32_16x16x4_f32(false, a1, false, b2, (short)0, acc[1][2], false, false);
        acc[1][3] = __builtin_amdgcn_wmma_f32_16x16x4_f32(false, a1, false, b3, (short)0, acc[1][3], false, false);
    }

    // ---- Epilogue: D layout = lanes carry N (co), VGPR v carries M (ow):
    //      lane owns (co = nt*16+mx), m = v + 8*hi -> 8 consecutive ow. ----
#pragma unroll
    for (int t = 0; t < 2; ++t) {
        const int ohv = oh0 + oy + t;
#pragma unroll
        for (int nt = 0; nt < 4; ++nt) {
            const int co = nt * 16 + mx;
            const float bb = lds[BLDS_OFF + co];
            float* op = out + (((size_t)(bat * CO + co) * OH + ohv) * OW)
                            + ow0 + dxw + 8 * hi;
            v8f a = acc[t][nt];
            float4 lo, hq;
            lo.x = silu(a[0] + bb); lo.y = silu(a[1] + bb);
            lo.z = silu(a[2] + bb); lo.w = silu(a[3] + bb);
            hq.x = silu(a[4] + bb); hq.y = silu(a[5] + bb);
            hq.z = silu(a[6] + bb); hq.w = silu(a[7] + bb);
            *(float4*)(op)     = lo;
            *(float4*)(op + 4) = hq;
        }
    }
}

extern "C" void kernel_launch(void* const* d_in, const int* in_sizes, int n_in,
                              void* d_out, int out_size, void* d_ws, size_t ws_size,
                              hipStream_t stream) {
    const float* x  = (const float*)d_in[0];   // (16,3,640,640)
    const float* W  = (const float*)d_in[1];   // (64,12,3,3)
    const float* b  = (const float*)d_in[2];   // (64)
    float* out      = (float*)d_out;           // (16,64,320,320)

    dim3 grid(Bb * (OH / 8) * (OW / 32));      // 16*40*10 = 6400
    dim3 block(256);                           // 8 waves of 32
    GConvFocus_18167711662710_kernel<<<grid, block, 0, stream>>>(x, W, b, out);
}